// HyperbolicPromptPool_7516192768899
// MI455X (gfx1250) — compile-verified
//
#include <hip/hip_runtime.h>
#include <hip/hip_bf16.h>
#include <math.h>

// ---------------- problem constants ----------------
#define BSZ   1024
#define LSEQ  197
#define DIM   768
#define POOLN 600
#define PLEN  5
#define TOPK  5
#define OUTL  (TOPK*PLEN + LSEQ)   // 222
#define EPSF  1e-12f
#define MAXNORM 0.99999f           // (1-1e-5)/sqrt(c), c=1

// d_out flat layout (floats), reference return order
static const size_t N_PROMPTED = (size_t)BSZ * OUTL * DIM;   // 174,587,904
static const size_t OFF_SIM    = N_PROMPTED;
static const size_t N_SIM      = (size_t)BSZ * POOLN;        // 614,400
static const size_t OFF_RSIM   = OFF_SIM + N_SIM;
static const size_t OFF_SKEY   = OFF_RSIM + 1;
static const size_t N_SKEY     = (size_t)BSZ * TOPK * DIM;   // 3,932,160
static const size_t OFF_IDX    = OFF_SKEY + N_SKEY;          // 5,120 elems

// workspace layout (float elements)
static const size_t WQ    = 0;                               // q      [1024,768]
static const size_t WQB   = WQ   + (size_t)BSZ  * DIM;       // q_ball [1024,768]
static const size_t WKB   = WQB  + (size_t)BSZ  * DIM;       // k_ball [600,768]
static const size_t WX2   = WKB  + (size_t)POOLN* DIM;       // x2 [1024]
static const size_t WY2   = WX2  + BSZ;                      // y2 [600]
static const size_t WRSUM = WY2  + POOLN;                    // row dist sums [1024]
static const size_t WIDX  = WRSUM+ BSZ;                      // int idx [1024*5]

typedef __attribute__((ext_vector_type(2))) float v2f;
typedef __attribute__((ext_vector_type(8))) float v8f;

// ------------------------------------------------------------------
// Kernel 1: q = mean_l(x_embed)  AND copy x_embed -> prompted[:,25:,:]
// Single pass over the 620MB input at b128 granularity: 192 threads,
// thread t owns float4 column t of the 768-wide rows. Per-thread
// float4 accumulator -> no cross-thread reduction needed.
// ------------------------------------------------------------------
__global__ __launch_bounds__(192) void mean_copy_kernel(
    const float4* __restrict__ x4, float4* __restrict__ prompted4,
    float4* __restrict__ q4) {
  const int b = blockIdx.x;
  const int t = threadIdx.x;                       // 0..191 (DIM/4 = 192)
  const int W = DIM / 4;                           // 192 float4 per row
  const float4* xb = x4 + (size_t)b * LSEQ * W;
  float4* pb = prompted4 + (size_t)b * OUTL * W + (size_t)(TOPK * PLEN) * W;
  float4 s = make_float4(0.f, 0.f, 0.f, 0.f);
  for (int l = 0; l < LSEQ; ++l) {
    const float4 v = xb[(size_t)l * W + t];        // global_load_b128
    s.x += v.x; s.y += v.y; s.z += v.z; s.w += v.w;
    pb[(size_t)l * W + t] = v;                     // global_store_b128
  }
  const float inv = 1.0f / (float)LSEQ;
  s.x *= inv; s.y *= inv; s.z *= inv; s.w *= inv;
  q4[(size_t)b * W + t] = s;
}

// ------------------------------------------------------------------
// Kernel 2: map_to_ball rows of q (blocks 0..1023) and prompt_key
// (blocks 1024..1623). Stores ball vector + squared ball-norm.
// ------------------------------------------------------------------
__global__ __launch_bounds__(256) void map_ball_kernel(
    const float* __restrict__ q, const float* __restrict__ pkey,
    float* __restrict__ qb, float* __restrict__ kb,
    float* __restrict__ x2, float* __restrict__ y2) {
  const int r = blockIdx.x;
  const int t = threadIdx.x;
  const bool isQ = (r < BSZ);
  const float* src = isQ ? (q + (size_t)r * DIM) : (pkey + (size_t)(r - BSZ) * DIM);
  float*       dst = isQ ? (qb + (size_t)r * DIM) : (kb + (size_t)(r - BSZ) * DIM);

  float v0 = src[t], v1 = src[t + 256], v2 = src[t + 512];

  __shared__ float red[256];
  red[t] = v0 * v0 + v1 * v1 + v2 * v2;
  __syncthreads();
  for (int s = 128; s > 0; s >>= 1) {
    if (t < s) red[t] += red[t + s];
    __syncthreads();
  }
  const float sumsq = red[0];

  // l2_normalize then * SCALE(0.1)
  const float rin = rsqrtf(fmaxf(sumsq, EPSF));
  const float u0 = v0 * rin * 0.1f, u1 = v1 * rin * 0.1f, u2 = v2 * rin * 0.1f;
  // expmap0: tanh(||u||) * u / ||u||   (sqrt(c)=1)
  const float un2 = sumsq * rin * rin * 0.01f;
  const float un  = sqrtf(fmaxf(un2, EPSF));
  const float f   = tanhf(un) / un;
  float b0 = u0 * f, b1 = u1 * f, b2 = u2 * f;
  float bn = tanhf(un);
  // proju0 clamp (no-op here numerically, kept for fidelity)
  const float sc = (bn > MAXNORM) ? (MAXNORM / bn) : 1.0f;
  b0 *= sc; b1 *= sc; b2 *= sc; bn *= sc;

  dst[t] = b0; dst[t + 256] = b1; dst[t + 512] = b2;
  if (t == 0) {
    if (isQ) x2[r] = bn * bn; else y2[r - BSZ] = bn * bn;
  }
}

// ------------------------------------------------------------------
// Kernel 3: pairwise Poincare distance via WMMA f32 16x16x4.
// xy = q_ball @ k_ball^T, then closed-form ||num||^2, no D-dim diff.
// grid = (ceil(600/16)=38, 1024/16=64), 32 threads (one wave) / tile.
// ------------------------------------------------------------------
__global__ __launch_bounds__(32) void dist_wmma_kernel(
    const float* __restrict__ qb, const float* __restrict__ kb,
    const float* __restrict__ x2, const float* __restrict__ y2,
    float* __restrict__ sim) {
  const int n0   = blockIdx.x * 16;
  const int m0   = blockIdx.y * 16;
  const int lane = threadIdx.x;

  // f32 16x4 A-fragment layout: lanes 0-15 -> K{0,1}, lanes 16-31 -> K{2,3}
  const int row  = lane & 15;
  const int koff = (lane >> 4) << 1;
  const int nc   = n0 + (lane & 15);
  const int ncl  = (nc < POOLN) ? nc : (POOLN - 1);   // clamp loads on edge tile

  const float* aBase = qb + (size_t)(m0 + row) * DIM + koff;
  const float* bBase = kb + (size_t)ncl * DIM + koff;  // B[k][n] = k_ball[n][k]

  v8f c = {};
#pragma unroll 4
  for (int k0 = 0; k0 < DIM; k0 += 4) {
    v2f a = *reinterpret_cast<const v2f*>(aBase + k0);
    v2f b = *reinterpret_cast<const v2f*>(bBase + k0);
    c = __builtin_amdgcn_wmma_f32_16x16x4_f32(false, a, false, b, (short)0, c,
                                              false, false);
  }

  if (nc < POOLN) {
    const float y2v = y2[nc];
#pragma unroll
    for (int r = 0; r < 8; ++r) {
      const int m = m0 + r + ((lane >> 4) << 3);   // C/D layout: M = r (+8 hi half)
      const float x2v = x2[m];
      const float xy  = c[r];
      // num = -alpha*x + beta*y ; ||num||^2 closed form
      const float denom = fmaxf(1.0f - 2.0f * xy + x2v * y2v, EPSF);
      const float alpha = 1.0f - 2.0f * xy + y2v;
      const float beta  = 1.0f - x2v;
      const float num2 = alpha * alpha * x2v - 2.0f * alpha * beta * xy
                       + beta * beta * y2v;
      const float dn  = sqrtf(fmaxf(num2 / (denom * denom), EPSF));
      const float arg = fminf(dn, 1.0f - 1e-7f);
      const float dist = 2.0f * atanhf(arg);
      sim[(size_t)m * POOLN + nc] = -dist;
    }
  }
}

// ------------------------------------------------------------------
// Kernel 4: per-row top-5 of similarity, sort idx ascending,
// emit idx (float to d_out, int to ws) + per-row distance sum.
// One thread per row; row fits in L2.
// ------------------------------------------------------------------
__global__ __launch_bounds__(256) void topk_kernel(
    const float* __restrict__ sim, float* __restrict__ out_idx,
    int* __restrict__ idx_int, float* __restrict__ rowsum) {
  const int b = blockIdx.x * blockDim.x + threadIdx.x;
  if (b >= BSZ) return;
  const float* s = sim + (size_t)b * POOLN;
  float bv[TOPK]; int bi[TOPK];
#pragma unroll
  for (int k = 0; k < TOPK; ++k) { bv[k] = -INFINITY; bi[k] = -1; }
  for (int p = 0; p < POOLN; ++p) {
    const float v = s[p];
    if (v > bv[TOPK - 1]) {                  // strict: keeps earliest index on tie
      int k = TOPK - 1;
      while (k > 0 && bv[k - 1] < v) { bv[k] = bv[k - 1]; bi[k] = bi[k - 1]; --k; }
      bv[k] = v; bi[k] = p;
    }
  }
  float dsum = 0.f;
#pragma unroll
  for (int k = 0; k < TOPK; ++k) dsum += -bv[k];   // distance = -similarity
  // sort indices ascending (5-element insertion sort)
#pragma unroll
  for (int i = 1; i < TOPK; ++i) {
    int key = bi[i]; int j = i - 1;
    while (j >= 0 && bi[j] > key) { bi[j + 1] = bi[j]; --j; }
    bi[j + 1] = key;
  }
#pragma unroll
  for (int k = 0; k < TOPK; ++k) {
    out_idx[(size_t)b * TOPK + k] = (float)bi[k];
    idx_int[b * TOPK + k] = bi[k];
  }
  rowsum[b] = dsum;
}

// ------------------------------------------------------------------
// Kernel 5: prompted[:, :25, :] = prompt[idx]   (float4 copies)
// grid = B*25 blocks x 192 threads
// ------------------------------------------------------------------
__global__ __launch_bounds__(192) void gather_prompt_kernel(
    const float* __restrict__ prompt, const int* __restrict__ idx_int,
    float* __restrict__ prompted) {
  const int j = blockIdx.x;
  const int b = j / (TOPK * PLEN);
  const int t = j % (TOPK * PLEN);
  const int k = t / PLEN, s = t % PLEN;
  const int pi = idx_int[b * TOPK + k];
  const float4* src = reinterpret_cast<const float4*>(
      prompt + ((size_t)pi * PLEN + s) * DIM);
  float4* dst = reinterpret_cast<float4*>(
      prompted + (size_t)b * OUTL * DIM + (size_t)t * DIM);
  dst[threadIdx.x] = src[threadIdx.x];
}

// ------------------------------------------------------------------
// Kernel 6: selected_key = k_ball[idx]  (scalar stores: output region
// is only 4-byte aligned because it sits after the reduce_sim scalar)
// grid = B*5 blocks x 256 threads
// ------------------------------------------------------------------
__global__ __launch_bounds__(256) void gather_key_kernel(
    const float* __restrict__ kb, const int* __restrict__ idx_int,
    float* __restrict__ skey) {
  const int j = blockIdx.x;
  const int b = j / TOPK, k = j % TOPK;
  const int pi = idx_int[b * TOPK + k];
  const float* src = kb + (size_t)pi * DIM;
  float* dst = skey + ((size_t)b * TOPK + k) * DIM;
  const int t = threadIdx.x;
  dst[t] = src[t]; dst[t + 256] = src[t + 256]; dst[t + 512] = src[t + 512];
}

// ------------------------------------------------------------------
// Kernel 7: reduce_sim = sum(rowsum)/B   (deterministic tree reduce)
// ------------------------------------------------------------------
__global__ __launch_bounds__(256) void reduce_sim_kernel(
    const float* __restrict__ rowsum, float* __restrict__ out_rsim) {
  __shared__ float red[256];
  const int t = threadIdx.x;
  red[t] = rowsum[t] + rowsum[t + 256] + rowsum[t + 512] + rowsum[t + 768];
  __syncthreads();
  for (int s = 128; s > 0; s >>= 1) {
    if (t < s) red[t] += red[t + s];
    __syncthreads();
  }
  if (t == 0) out_rsim[0] = red[0] / (float)BSZ;
}

// ------------------------------------------------------------------
extern "C" void kernel_launch(void* const* d_in, const int* in_sizes, int n_in,
                              void* d_out, int out_size, void* d_ws, size_t ws_size,
                              hipStream_t stream) {
  const float* x_embed    = (const float*)d_in[0];
  const float* prompt     = (const float*)d_in[1];
  const float* prompt_key = (const float*)d_in[2];

  float* out = (float*)d_out;
  float* prompted = out;
  float* sim      = out + OFF_SIM;
  float* rsim     = out + OFF_RSIM;
  float* skey     = out + OFF_SKEY;
  float* oidx     = out + OFF_IDX;

  float* ws   = (float*)d_ws;
  float* q    = ws + WQ;
  float* qb   = ws + WQB;
  float* kb   = ws + WKB;
  float* x2   = ws + WX2;
  float* y2   = ws + WY2;
  float* rsum = ws + WRSUM;
  int*   idxi = (int*)(ws + WIDX);

  // 1) fused mean + bulk copy, b128 per lane (single pass over x_embed)
  mean_copy_kernel<<<BSZ, 192, 0, stream>>>(
      (const float4*)x_embed, (float4*)prompted, (float4*)q);
  // 2) map_to_ball for q rows and prompt_key rows
  map_ball_kernel<<<BSZ + POOLN, 256, 0, stream>>>(q, prompt_key, qb, kb, x2, y2);
  // 3) WMMA pairwise Poincare distances -> similarity
  dist_wmma_kernel<<<dim3((POOLN + 15) / 16, BSZ / 16), 32, 0, stream>>>(
      qb, kb, x2, y2, sim);
  // 4) top-5 per row
  topk_kernel<<<BSZ / 256, 256, 0, stream>>>(sim, oidx, idxi, rsum);
  // 5) gather prompts into prompted[:, :25, :]
  gather_prompt_kernel<<<BSZ * TOPK * PLEN, 192, 0, stream>>>(prompt, idxi, prompted);
  // 6) gather selected keys
  gather_key_kernel<<<BSZ * TOPK, 256, 0, stream>>>(kb, idxi, skey);
  // 7) deterministic reduce_sim
  reduce_sim_kernel<<<1, 256, 0, stream>>>(rsum, rsim);
}